// VolumetricCELoss_146028888122
// MI455X (gfx1250) — compile-verified
//
#include <hip/hip_runtime.h>
#include <hip/hip_bf16.h>

typedef __attribute__((ext_vector_type(2))) float v2f;
typedef __attribute__((ext_vector_type(8))) float v8f;

#define B_            8
#define J_            17
#define V3_           (64*64*64)              // 262144 voxels per batch
#define TILES_        (V3_/16)                // 16384 16-voxel tiles per batch
#define BLOCKS_PER_B  64
#define WAVES_PER_BLK 8
#define TILES_PER_BLK (TILES_/BLOCKS_PER_B)   // 256
#define TILES_PER_WAVE (TILES_PER_BLK/WAVES_PER_BLK) // 32 (processed 2 at a time)
#define SLOTS_PER_B   32

// ---------------------------------------------------------------- init
__global__ void vce_init_ws(unsigned long long* ws) {
    int i = blockIdx.x * blockDim.x + threadIdx.x;
    if (i < B_ * SLOTS_PER_B) ws[i] = 0xFFFFFFFFFFFFFFFFull;
}

// monotone float -> u32 key (ascending float order == ascending unsigned order)
__device__ __forceinline__ unsigned int fkey(float f) {
    unsigned int b = __float_as_uint(f);
    unsigned int mask = (unsigned int)(((int)b) >> 31) | 0x80000000u;
    return b ^ mask;
}

__device__ __forceinline__ unsigned long long pack_min(float v, unsigned int idx) {
    return ((unsigned long long)fkey(v) << 32) | (unsigned long long)idx;
}

// ---------------------------------------------------------------- argmin via WMMA
// score[m][n] = |c_m|^2 - 2 c_m . k_n  =  A(16x4) x B(4x16):
//   A row m = (cx, cy, cz, |c|^2),  B col n = (-2kx, -2ky, -2kz, 1)
// Joints 0..15 via one V_WMMA_F32_16X16X4_F32 per voxel tile; joint 16 via 3 FMAs/lane.
__global__ void __launch_bounds__(256)
vce_argmin_kernel(const float* __restrict__ coord,  // (B, V3, 3)
                  const float* __restrict__ kp,     // (B, J, 3)
                  unsigned long long* __restrict__ ws) {
    const int b    = blockIdx.x / BLOCKS_PER_B;
    const int blk  = blockIdx.x % BLOCKS_PER_B;
    const int wave = threadIdx.x >> 5;
    const int lane = threadIdx.x & 31;
    const int l15  = lane & 15;
    const bool hi  = lane >= 16;              // upper half: K=2,3 of A; rows 8..15 of D

    const float* cbase = coord + (size_t)b * V3_ * 3;
    const float* kpb   = kp    + (size_t)b * J_ * 3;

    // B operand: joints 0..15
    float k0x = kpb[l15*3 + 0];
    float k0y = kpb[l15*3 + 1];
    float k0z = kpb[l15*3 + 2];
    v2f b0;
    b0.x = hi ? (-2.0f*k0z) : (-2.0f*k0x);
    b0.y = hi ? 1.0f        : (-2.0f*k0y);

    // joint 16 constants (uniform -> scalar loads)
    const float m2kx = -2.0f * kp[(size_t)b*J_*3 + 16*3 + 0];
    const float m2ky = -2.0f * kp[(size_t)b*J_*3 + 16*3 + 1];
    const float m2kz = -2.0f * kp[(size_t)b*J_*3 + 16*3 + 2];

    const float INF = __int_as_float(0x7F800000);
    float m0v = INF;  unsigned int m0i = 0u;   // tile-joint l15 (joints 0..15)
    float m2v = INF;  unsigned int m2i = 0u;   // joint 16

    const v8f zero = {};
    const int tbase = blk * TILES_PER_BLK + wave * TILES_PER_WAVE;

    for (int t = 0; t < TILES_PER_WAVE; t += 2) {
        const int n0 = (tbase + t) * 16;
        const int gA = n0 + l15;
        const int gB = gA + 16;
        __builtin_prefetch(cbase + 3*(gA + 32), 0, 0);   // global_prefetch_b8
        __builtin_prefetch(cbase + 3*(gB + 32), 0, 0);

        // two independent b96 loads -> overlapped latency
        float axc = cbase[3*gA + 0], ayc = cbase[3*gA + 1], azc = cbase[3*gA + 2];
        float bxc = cbase[3*gB + 0], byc = cbase[3*gB + 1], bzc = cbase[3*gB + 2];

        float an2 = axc*axc + ayc*ayc + azc*azc;
        float bn2 = bxc*bxc + byc*byc + bzc*bzc;

        v2f aA, aB;                    // A 16x4 layout: VGPR0={K0|K2}, VGPR1={K1|K3}
        aA.x = hi ? azc : axc;  aA.y = hi ? an2 : ayc;
        aB.x = hi ? bzc : bxc;  aB.y = hi ? bn2 : byc;

        v8f d0 = __builtin_amdgcn_wmma_f32_16x16x4_f32(
                     false, aA, false, b0, (short)0, zero, false, false);
        v8f d1 = __builtin_amdgcn_wmma_f32_16x16x4_f32(
                     false, aB, false, b0, (short)0, zero, false, false);

        // joint 16 in VALU (3 fma each); idx = this lane's own voxel
        float sA = fmaf(axc, m2kx, fmaf(ayc, m2ky, fmaf(azc, m2kz, an2)));
        float sB = fmaf(bxc, m2kx, fmaf(byc, m2ky, fmaf(bzc, m2kz, bn2)));
        if (sA < m2v) { m2v = sA; m2i = (unsigned int)gA; }
        if (sB < m2v) { m2v = sB; m2i = (unsigned int)gB; }

        // tile argmin updates: lane holds col n=l15, rows m = r + 8*(lane/16)
        const unsigned int rbA = (unsigned int)n0 + (hi ? 8u : 0u);
        const unsigned int rbB = rbA + 16u;
        #pragma unroll
        for (int r = 0; r < 8; ++r) {
            float s0 = d0[r];
            if (s0 < m0v) { m0v = s0; m0i = rbA + (unsigned int)r; }
        }
        #pragma unroll
        for (int r = 0; r < 8; ++r) {
            float s1 = d1[r];
            if (s1 < m0v) { m0v = s1; m0i = rbB + (unsigned int)r; }
        }
    }

    // ---- combine partners & wave-reduce joint 16, then one atomic each ----
    unsigned long long p0 = pack_min(m0v, m0i);
    {   // lanes l and l+16 share the same joint (different row subsets)
        unsigned long long q = __shfl_xor(p0, 16, 32);
        p0 = (q < p0) ? q : p0;
    }
    unsigned long long p1 = pack_min(m2v, m2i);
    #pragma unroll
    for (int off = 16; off > 0; off >>= 1) {
        unsigned long long q = __shfl_xor(p1, off, 32);
        p1 = (q < p1) ? q : p1;
    }
    if (lane < 16) atomicMin(&ws[b * SLOTS_PER_B + l15], p0);
    if (lane == 0) atomicMin(&ws[b * SLOTS_PER_B + 16],  p1);
}

// ---------------------------------------------------------------- finalize
__global__ void __launch_bounds__(256)
vce_finalize_kernel(const unsigned long long* __restrict__ ws,
                    const float* __restrict__ pred,      // (B, J, V3)
                    const float* __restrict__ validity,  // (B, J, 1)
                    float* __restrict__ out) {
    __shared__ float red[256];
    const int t = threadIdx.x;
    float term = 0.f;
    if (t < B_ * J_) {
        const int b = t / J_, j = t % J_;
        unsigned long long p = ws[b * SLOTS_PER_B + j];
        unsigned int idx = (unsigned int)(p & 0xFFFFFFFFull);
        float pv = pred[((size_t)b * J_ + j) * V3_ + idx];
        term = validity[t] * (-logf(pv + 1e-6f));
    }
    red[t] = term;
    __syncthreads();
    for (int s = 128; s > 0; s >>= 1) {
        if (t < s) red[t] += red[t + s];
        __syncthreads();
    }
    if (t == 0) out[0] = red[0] / (float)(B_ * J_);
}

// ---------------------------------------------------------------- launch
extern "C" void kernel_launch(void* const* d_in, const int* in_sizes, int n_in,
                              void* d_out, int out_size, void* d_ws, size_t ws_size,
                              hipStream_t stream) {
    const float* coord = (const float*)d_in[0];   // (8,64,64,64,3) f32
    const float* pred  = (const float*)d_in[1];   // (8,17,64,64,64) f32
    const float* kp    = (const float*)d_in[2];   // (8,17,3) f32
    const float* valid = (const float*)d_in[3];   // (8,17,1) f32
    unsigned long long* ws = (unsigned long long*)d_ws;

    hipLaunchKernelGGL(vce_init_ws, dim3(1), dim3(256), 0, stream, ws);
    hipLaunchKernelGGL(vce_argmin_kernel, dim3(B_ * BLOCKS_PER_B), dim3(256), 0, stream,
                       coord, kp, ws);
    hipLaunchKernelGGL(vce_finalize_kernel, dim3(1), dim3(256), 0, stream,
                       ws, pred, valid, (float*)d_out);
}